// ProposedConv_81269371175022
// MI455X (gfx1250) — compile-verified
//
#include <hip/hip_runtime.h>
#include <math.h>

// Problem constants (fixed by the reference).
#define N_NODES 8192
#define N_EDGES 4096
#define DIN     256
#define DOUT    128
#define NNZV    65536
#define EDGE_CAP 64   // Poisson(16) degree; P(>64) ~ 1e-18
#define NODE_CAP 64   // Poisson(8)  degree

constexpr float kE1   = 2.71828182845904523536f; // e^1 (softmax over {0,1} rows)
constexpr float kTemp = 11.3137084989847603904f; // sqrt(DOUT)

typedef __attribute__((ext_vector_type(2))) float    v2f;
typedef __attribute__((ext_vector_type(8))) float    v8f;
typedef __attribute__((ext_vector_type(4))) unsigned v4u;
typedef __attribute__((ext_vector_type(8))) unsigned v8u;

// ---------------------------------------------------------------- utilities
__global__ void k_zero(float* p, size_t n) {
  size_t i = (size_t)blockIdx.x * blockDim.x + threadIdx.x;
  size_t st = (size_t)gridDim.x * blockDim.x;
  for (; i < n; i += st) p[i] = 0.f;
}

// Build incidence bitmaps; atomicOr dedupes duplicate (node,edge) pairs,
// exactly matching jnp .at[].set(1.0).
__global__ void k_scatter(const int* __restrict__ he,
                          unsigned* __restrict__ be, unsigned* __restrict__ bn) {
  int i = blockIdx.x * blockDim.x + threadIdx.x;
  if (i >= NNZV) return;
  int n = he[i];
  int e = he[NNZV + i];
  atomicOr(&be[(size_t)e * (N_NODES / 32) + (n >> 5)], 1u << (n & 31));
  atomicOr(&bn[(size_t)n * (N_EDGES / 32) + (e >> 5)], 1u << (e & 31));
}

// --------------------------------------------------- fp32 WMMA GEMM (16x16x4)
// C[M,128] = A[M,K] @ B[K,128] (+bias). One block (8 waves) per 16-row strip:
// the 16xK A strip is DMA'd to LDS once by the Tensor Data Mover with a
// 4-dword per-row pad (bank-conflict-free reads), then wave w computes the
// 16x16 output tile for columns [16w,16w+16) with V_WMMA_F32_16X16X4_F32.
// A-frag layout (ISA 7.12.2, 32-bit A 16x4): lane&15 = row; lanes<16 hold
// K={k,k+1}, lanes>=16 hold K={k+2,k+3}. B/C/D: row striped across lanes.
__global__ void k_gemm_wmma(const float* __restrict__ A, const float* __restrict__ B,
                            const float* __restrict__ bias, float* __restrict__ C,
                            int K, int NO) {
  __shared__ float sA[16 * (256 + 4)];   // sole LDS object -> offset 0
  int tm = blockIdx.x;
  int tid = threadIdx.x;                 // 256 threads = 8 waves
  int wave = tid >> 5;
  int lane = tid & 31;
  int KP = K + 4;                        // padded row stride (dwords)

  if (wave == 0) {
    // ---- Tensor DMA descriptor (D#), ISA ch.8 ----
    unsigned long long ga = (unsigned long long)(const void*)(A + (size_t)tm * 16 * K);
    unsigned pic = (K == 256) ? 7u : 6u; // pad_interval code: 2<<pic dwords = K
    v4u g0;
    g0[0] = 1u;                                               // count=1, user mode
    g0[1] = 0u;                                               // lds_addr (sA @ 0)
    g0[2] = (unsigned)ga;                                     // global_addr[31:0]
    g0[3] = ((unsigned)(ga >> 32) & 0x01FFFFFFu) | 0x80000000u; // [56:32] | type=2
    unsigned td0 = (unsigned)K;          // tensor_dim0 (elements)
    unsigned td1 = gridDim.x * 16u;      // tensor_dim1 (rows)
    v8u g1;
    g1[0] = 0x20000u                     // data_size=2 (4B), wg_mask=0
          | (1u << 20)                   // pad_enable
          | (pic << 22)                  // pad_interval: one row
          | (3u << 25);                  // pad_amount: 4 dwords
    g1[1] = (td0 & 0xFFFFu) << 16;                         // dim0[15:0] @ bits63:48
    g1[2] = (td0 >> 16) | ((td1 & 0xFFFFu) << 16);         // dim0[31:16], dim1[15:0]
    g1[3] = (td1 >> 16) | ((unsigned)(K & 0xFFFF) << 16);  // dim1[31:16], tile_dim0=K
    g1[4] = 16u;                                           // tile_dim1=16, tile_dim2=0
    g1[5] = (unsigned)K;                                   // dim0_stride[31:0]
    g1[6] = 0u;                                            // dim0_stride[47:32], dim1_stride lo
    g1[7] = 0u;                                            // dim1_stride hi
    // The LDS base pointer is passed as an operand so its address ESCAPES into
    // the asm: without this, LLVM proves the internal LDS array is never
    // stored-to and folds all sA reads to undef (observed in round 2: ds count
    // unchanged, WMMA A-operand loop-invariant). With the escape + "memory"
    // clobber, the TDM write is visible and the ds_load_b64 A-fragment reads
    // must be materialized.
    asm volatile("tensor_load_to_lds %0, %1 ; lds dst %2"
                 :: "s"(g0), "s"(g1), "v"(&sA[0]) : "memory");
    __builtin_amdgcn_s_wait_tensorcnt(0);
  }
  __syncthreads();

  bool hi = lane >= 16;
  int col = lane & 15;
  int tn = wave;                         // 8 waves x 16 cols = 128 = DOUT
  const float* Bcol = B + tn * 16 + col;
  const float* Ar = sA + (size_t)col * KP;  // row (lane&15) of the A strip
  v8f acc = {};
  for (int k = 0; k < K; k += 4) {
    int ka = k + (hi ? 2 : 0);
    v2f av; av[0] = Ar[ka];               av[1] = Ar[ka + 1];
    v2f bv; bv[0] = Bcol[(size_t)ka * NO];  bv[1] = Bcol[(size_t)(ka + 1) * NO];
    acc = __builtin_amdgcn_wmma_f32_16x16x4_f32(
        false, av, false, bv, (short)0, acc, false, false);
  }
  int row0 = tm * 16 + (hi ? 8 : 0);
  float badd = bias ? bias[tn * 16 + col] : 0.f;
  for (int r = 0; r < 8; ++r)
    C[(size_t)(row0 + r) * NO + tn * 16 + col] = acc[r] + badd;
}

// row[n] = dot(q_ctx, x4[n]) / sqrt(D)
__global__ void k_row(const float* __restrict__ x4, const float* __restrict__ qc,
                      float* __restrict__ rowv) {
  int i = blockIdx.x * blockDim.x + threadIdx.x;
  if (i >= N_NODES) return;
  float s = 0.f;
  for (int d = 0; d < DOUT; ++d) s += qc[d] * x4[(size_t)i * DOUT + d];
  rowv[i] = s / kTemp;
}

// column sums of xt and x4 (the rank-1 "u_e * 1^T" terms)
__global__ void k_colsum(const float* __restrict__ xt, const float* __restrict__ x4,
                         float* __restrict__ cs_xt, float* __restrict__ cs_x4) {
  int d = threadIdx.x;            // 0..127
  int b = blockIdx.x;             // 64 blocks x 128 rows
  float a = 0.f, c = 0.f;
  for (int r = b * 128; r < (b + 1) * 128; ++r) {
    a += xt[(size_t)r * DOUT + d];
    c += x4[(size_t)r * DOUT + d];
  }
  atomicAdd(&cs_xt[d], a);
  atomicAdd(&cs_x4[d], c);
}

// ------------------------------------------------- attention 1 (per edge)
__global__ void k_edge_attn(const unsigned* __restrict__ be, const float* __restrict__ rowv,
                            const float* __restrict__ xt, const float* __restrict__ x4,
                            const float* __restrict__ cs_xt, const float* __restrict__ cs_x4,
                            float* __restrict__ edgef, float* __restrict__ G1,
                            float* __restrict__ v1,
                            int* __restrict__ emem, float* __restrict__ ecoef,
                            int* __restrict__ edeg) {
  __shared__ int s_cnt[128];
  __shared__ int s_mem[EDGE_CAP];
  __shared__ float s_val[EDGE_CAP];
  __shared__ float s_coef[EDGE_CAP];
  __shared__ float s_sum;
  int e = blockIdx.x, t = threadIdx.x; // 128 threads
  const unsigned* br = be + (size_t)e * (N_NODES / 32);
  unsigned w0 = br[2 * t], w1 = br[2 * t + 1];
  int c = __popc(w0) + __popc(w1);
  s_cnt[t] = c; __syncthreads();
  for (int s = 1; s < 128; s <<= 1) {          // deterministic inclusive scan
    int v = s_cnt[t]; int add = (t >= s) ? s_cnt[t - s] : 0;
    __syncthreads(); s_cnt[t] = v + add; __syncthreads();
  }
  int excl = s_cnt[t] - c;
  int dtot = s_cnt[127];
  { int idx = excl; unsigned b = w0; int base = (2 * t) * 32;
    while (b) { int bit = __ffs(b) - 1; if (idx < EDGE_CAP) s_mem[idx] = base + bit; ++idx; b &= b - 1; }
    b = w1; base = (2 * t + 1) * 32;
    while (b) { int bit = __ffs(b) - 1; if (idx < EDGE_CAP) s_mem[idx] = base + bit; ++idx; b &= b - 1; } }
  __syncthreads();
  int deg = dtot < EDGE_CAP ? dtot : EDGE_CAP;
  float denom = (float)dtot * kE1 + (float)(N_NODES - dtot);
  float u = 1.f / denom;
  float ueff = (dtot == 0) ? (u + 1.f / (float)N_NODES) : u; // deg-0: uniform fold-in
  float ce = (kE1 - 1.f) / denom;
  if (t < deg) s_val[t] = rowv[s_mem[t]];
  __syncthreads();
  if (t == 0 && deg > 0) {
    float m = -3.0e38f;
    for (int j = 0; j < deg; ++j) m = fmaxf(m, s_val[j]);
    float s = 0.f;
    for (int j = 0; j < deg; ++j) { float p = expf(s_val[j] - m); s_val[j] = p; s += p; }
    s_sum = s;
  }
  __syncthreads();
  if (t < deg) s_coef[t] = ce + s_val[t] / s_sum;   // c_e + p1[e,n]
  __syncthreads();
  if (t == 0) edeg[e] = deg;
  if (t < deg) { emem[(size_t)e * EDGE_CAP + t] = s_mem[t];
                 ecoef[(size_t)e * EDGE_CAP + t] = s_coef[t]; }
  float acc_e = ueff * cs_xt[t];   // edge = att1 @ xt
  float acc_g = ueff * cs_x4[t];   // G1   = att1 @ x4
  for (int j = 0; j < deg; ++j) {
    int nn = s_mem[j]; float cf = s_coef[j];
    acc_e += cf * xt[(size_t)nn * DOUT + t];
    acc_g += cf * x4[(size_t)nn * DOUT + t];
  }
  edgef[(size_t)e * DOUT + t] = acc_e;
  G1[(size_t)e * DOUT + t] = acc_g;
  atomicAdd(&v1[t], ueff * acc_g); // v1 = sum_e u_eff * G1[e]
}

// ------------------------------------------------- attention 2 (per node)
__global__ void k_node_attn(const unsigned* __restrict__ bn, const float* __restrict__ x4,
                            const float* __restrict__ edge4,
                            const int* __restrict__ emem, const float* __restrict__ ecoef,
                            const int* __restrict__ edeg,
                            int* __restrict__ nmem, float* __restrict__ ncoef,
                            float* __restrict__ nt1, int* __restrict__ ndeg,
                            float* __restrict__ node_u, float* __restrict__ wvec) {
  __shared__ int s_cnt[128];
  __shared__ int s_mem[NODE_CAP];
  __shared__ float s_val[NODE_CAP];
  __shared__ float s_coef[NODE_CAP];
  __shared__ float red[128];
  __shared__ float s_sum;
  int n = blockIdx.x, t = threadIdx.x;
  unsigned wbits = bn[(size_t)n * (N_EDGES / 32) + t];
  int c = __popc(wbits);
  s_cnt[t] = c; __syncthreads();
  for (int s = 1; s < 128; s <<= 1) {
    int v = s_cnt[t]; int add = (t >= s) ? s_cnt[t - s] : 0;
    __syncthreads(); s_cnt[t] = v + add; __syncthreads();
  }
  int excl = s_cnt[t] - c;
  int dtot = s_cnt[127];
  { int idx = excl; unsigned b = wbits; int base = t * 32;
    while (b) { int bit = __ffs(b) - 1; if (idx < NODE_CAP) s_mem[idx] = base + bit; ++idx; b &= b - 1; } }
  __syncthreads();
  int deg = dtot < NODE_CAP ? dtot : NODE_CAP;
  float denom = (float)dtot * kE1 + (float)(N_EDGES - dtot);
  float up = 1.f / denom;
  float upeff = (dtot == 0) ? (up + 1.f / (float)N_EDGES) : up;
  float cp = (kE1 - 1.f) / denom;
  float xv = x4[(size_t)n * DOUT + t];
  for (int j = 0; j < deg; ++j) {            // scores2 = dot(x4[n], edge4[e])/T
    red[t] = xv * edge4[(size_t)s_mem[j] * DOUT + t];
    __syncthreads();
    for (int s = 64; s > 0; s >>= 1) { if (t < s) red[t] += red[t + s]; __syncthreads(); }
    if (t == 0) s_val[j] = red[0] / kTemp;
    __syncthreads();
  }
  if (t == 0 && deg > 0) {
    float m = -3.0e38f;
    for (int j = 0; j < deg; ++j) m = fmaxf(m, s_val[j]);
    float s = 0.f;
    for (int j = 0; j < deg; ++j) { float p = expf(s_val[j] - m); s_val[j] = p; s += p; }
    s_sum = s;
  }
  __syncthreads();
  if (t < deg) s_coef[t] = cp + s_val[t] / s_sum;   // c'_n + p2[n,e]
  __syncthreads();
  if (t == 0) { ndeg[n] = deg; node_u[n] = upeff; }
  if (t < deg) {
    int ej = s_mem[t];
    nmem[(size_t)n * NODE_CAP + t] = ej;
    ncoef[(size_t)n * NODE_CAP + t] = s_coef[t];
    // T1 coefficient: locate n inside edge ej's member list -> c_e + p1[ej,n]
    int de = edeg[ej]; float cf = 0.f;
    const int* em = emem + (size_t)ej * EDGE_CAP;
    for (int k = 0; k < de; ++k)
      if (em[k] == n) { cf = ecoef[(size_t)ej * EDGE_CAP + k]; break; }
    nt1[(size_t)n * NODE_CAP + t] = cf;
  }
  atomicAdd(&wvec[t], upeff * xv);  // w = sum_n u'_eff * x4[n]
}

// G2[e] = w + sum_{n in e} (c'_n + p2[n,e]) * x4[n]  (edge-side, deterministic)
__global__ void k_g2(const int* __restrict__ emem, const int* __restrict__ edeg,
                     const int* __restrict__ nmem, const float* __restrict__ ncoef,
                     const int* __restrict__ ndeg, const float* __restrict__ x4,
                     const float* __restrict__ wvec,
                     float* __restrict__ G2, float* __restrict__ sumG2) {
  __shared__ int s_mem[EDGE_CAP];
  __shared__ float s_coef[EDGE_CAP];
  int e = blockIdx.x, t = threadIdx.x;
  int deg = edeg[e];
  if (t < deg) s_mem[t] = emem[(size_t)e * EDGE_CAP + t];
  __syncthreads();
  if (t < deg) {
    int nn = s_mem[t]; int dn = ndeg[nn]; float cf = 0.f;
    const int* nm = nmem + (size_t)nn * NODE_CAP;
    for (int k = 0; k < dn; ++k)
      if (nm[k] == e) { cf = ncoef[(size_t)nn * NODE_CAP + k]; break; }
    s_coef[t] = cf;
  }
  __syncthreads();
  float acc = wvec[t];
  for (int j = 0; j < deg; ++j) acc += s_coef[j] * x4[(size_t)s_mem[j] * DOUT + t];
  G2[(size_t)e * DOUT + t] = acc;
  atomicAdd(&sumG2[t], acc);
}

// h[n] = elu( v1 + u'_n*sumG2 + sum_{e in n} [ (c_e+p1)G1[e] + (c'_n+p2)G2[e] ] )
__global__ void k_final(const float* __restrict__ v1, const float* __restrict__ sumG2,
                        const float* __restrict__ node_u,
                        const int* __restrict__ nmem, const float* __restrict__ ncoef,
                        const float* __restrict__ nt1, const int* __restrict__ ndeg,
                        const float* __restrict__ G1, const float* __restrict__ G2,
                        float* __restrict__ out) {
  int n = blockIdx.x, t = threadIdx.x;
  int dn = ndeg[n];
  float up = node_u[n];
  float acc = v1[t] + up * sumG2[t];
  for (int j = 0; j < dn; ++j) {
    int ej = nmem[(size_t)n * NODE_CAP + j];
    acc += ncoef[(size_t)n * NODE_CAP + j] * G2[(size_t)ej * DOUT + t]
         + nt1[(size_t)n * NODE_CAP + j] * G1[(size_t)ej * DOUT + t];
  }
  out[(size_t)n * DOUT + t] = acc > 0.f ? acc : expm1f(acc);
}

// ---------------------------------------------------------------- launcher
extern "C" void kernel_launch(void* const* d_in, const int* in_sizes, int n_in,
                              void* d_out, int out_size, void* d_ws, size_t ws_size,
                              hipStream_t stream) {
  (void)in_sizes; (void)n_in; (void)out_size; (void)ws_size;
  const float* x    = (const float*)d_in[0];
  const float* w1   = (const float*)d_in[1];
  const float* w2   = (const float*)d_in[2];
  const float* w3   = (const float*)d_in[3];
  const float* bias = (const float*)d_in[4];
  const float* qc   = (const float*)d_in[5];
  const int*   he   = (const int*)d_in[6];
  float* out = (float*)d_out;

  char* base = (char*)d_ws;
  size_t off = 0;
  auto carve = [&](size_t bytes) -> char* {
    char* r = base + off;
    off = (off + bytes + 255) & ~(size_t)255;
    return r;
  };
  // ---- zero-init region (contiguous at ws start) ----
  unsigned* Bedge = (unsigned*)carve((size_t)N_EDGES * (N_NODES / 32) * 4);
  unsigned* Bnode = (unsigned*)carve((size_t)N_NODES * (N_EDGES / 32) * 4);
  float* v1    = (float*)carve(DOUT * 4);
  float* wvec  = (float*)carve(DOUT * 4);
  float* sG2   = (float*)carve(DOUT * 4);
  float* cs_xt = (float*)carve(DOUT * 4);
  float* cs_x4 = (float*)carve(DOUT * 4);
  size_t zero_words = off / 4;
  // ---- fully-overwritten buffers ----
  float* xt    = (float*)carve((size_t)N_NODES * DOUT * 4);
  float* x4    = (float*)carve((size_t)N_NODES * DOUT * 4);
  float* rowv  = (float*)carve((size_t)N_NODES * 4);
  float* edgef = (float*)carve((size_t)N_EDGES * DOUT * 4);
  float* G1    = (float*)carve((size_t)N_EDGES * DOUT * 4);
  float* edge4 = (float*)carve((size_t)N_EDGES * DOUT * 4);
  float* G2    = (float*)carve((size_t)N_EDGES * DOUT * 4);
  int*   emem  = (int*)carve((size_t)N_EDGES * EDGE_CAP * 4);
  float* ecoef = (float*)carve((size_t)N_EDGES * EDGE_CAP * 4);
  int*   edeg  = (int*)carve((size_t)N_EDGES * 4);
  int*   nmem  = (int*)carve((size_t)N_NODES * NODE_CAP * 4);
  float* ncoef = (float*)carve((size_t)N_NODES * NODE_CAP * 4);
  float* nt1   = (float*)carve((size_t)N_NODES * NODE_CAP * 4);
  int*   ndeg  = (int*)carve((size_t)N_NODES * 4);
  float* nodeu = (float*)carve((size_t)N_NODES * 4);

  k_zero<<<2048, 256, 0, stream>>>((float*)d_ws, zero_words);
  k_scatter<<<NNZV / 256, 256, 0, stream>>>(he, Bedge, Bnode);
  // xt = x@W + bias ; x4 = x@W2   (TDM-staged fp32 WMMA)
  k_gemm_wmma<<<N_NODES / 16, 256, 0, stream>>>(x, w1, bias, xt, DIN, DOUT);
  k_gemm_wmma<<<N_NODES / 16, 256, 0, stream>>>(x, w2, nullptr, x4, DIN, DOUT);
  k_row<<<N_NODES / 256, 256, 0, stream>>>(x4, qc, rowv);
  k_colsum<<<64, 128, 0, stream>>>(xt, x4, cs_xt, cs_x4);
  k_edge_attn<<<N_EDGES, 128, 0, stream>>>(Bedge, rowv, xt, x4, cs_xt, cs_x4,
                                           edgef, G1, v1, emem, ecoef, edeg);
  // edge4 = edge @ W3   (TDM-staged fp32 WMMA)
  k_gemm_wmma<<<N_EDGES / 16, 256, 0, stream>>>(edgef, w3, nullptr, edge4, DOUT, DOUT);
  k_node_attn<<<N_NODES, 128, 0, stream>>>(Bnode, x4, edge4, emem, ecoef, edeg,
                                           nmem, ncoef, nt1, ndeg, nodeu, wvec);
  k_g2<<<N_EDGES, 128, 0, stream>>>(emem, edeg, nmem, ncoef, ndeg, x4, wvec, G2, sG2);
  k_final<<<N_NODES, 128, 0, stream>>>(v1, sG2, nodeu, nmem, ncoef, nt1, ndeg, G1, G2, out);
}